// Net_87892210745629
// MI455X (gfx1250) — compile-verified
//
#include <hip/hip_runtime.h>
#include <math.h>
#include <stdint.h>

typedef float v2f __attribute__((ext_vector_type(2)));
typedef float v8f __attribute__((ext_vector_type(8)));

#define NCLI 10
#define TILE_K 512
#define GRAM_BLOCKS 2048

// -------- kernel 0: zero the 16x16 Gram accumulator (d_ws is poisoned) -----
__global__ void zero_gram(float* G) { G[threadIdx.x] = 0.0f; }

// Issue one 20 KB tile (TILE_K rows x 10 floats) as 5 async b128 loads/lane.
// IOFFSET is added to BOTH the global and LDS addresses and our chunk offsets
// match on both sides, so one LDS-addr VGPR + one 64b base covers all 5.
__device__ __forceinline__ void stage_async(unsigned lws, const float* gptr) {
    const uint64_t ga = (uint64_t)(uintptr_t)gptr;
    asm volatile(
        "global_load_async_to_lds_b128 %0, %1, off\n\t"
        "global_load_async_to_lds_b128 %0, %1, off offset:4096\n\t"
        "global_load_async_to_lds_b128 %0, %1, off offset:8192\n\t"
        "global_load_async_to_lds_b128 %0, %1, off offset:12288\n\t"
        "global_load_async_to_lds_b128 %0, %1, off offset:16384"
        :: "v"(lws), "v"(ga) : "memory");
}

// 16 chained V_WMMA_F32_16X16X4_F32 over one staged tile (64-wide K per wave).
// mc is an always-in-bounds column (lanes m>=10 re-read rows 4..9): the junk
// they produce lands only in C rows/cols >= 10, which are never consumed.
__device__ __forceinline__ void wmma_tile(const float* __restrict__ buf,
                                          int wave, int mc, int koff, v8f& acc) {
    const int kbase = wave * (TILE_K / 8);
#pragma unroll
    for (int kk = 0; kk < TILE_K / 8; kk += 4) {
        const int k = kbase + kk + koff;
        v2f a;
        a.x = buf[(k    ) * NCLI + mc];
        a.y = buf[(k + 1) * NCLI + mc];
        // D = A*B + C with A == B fragment (Gram symmetry)
        acc = __builtin_amdgcn_wmma_f32_16x16x4_f32(
            false, a, false, a, (short)0, acc, false, false);
    }
}

// -------- kernel 1: Gram = X * X^T via WMMA, double-buffered async LDS -----
// x is [d][10] row-major; logical A is [16][d]; only the 10x10 corner of the
// 16x16 accumulator is meaningful (padding rows carry finite junk by design).
__global__ __launch_bounds__(256) void gram_wmma(const float* __restrict__ x,
                                                 float* __restrict__ G, int d) {
    __shared__ float lds[2][TILE_K * NCLI];     // 2 x 20480 B
    const int tid  = threadIdx.x;
    const int lane = tid & 31;
    const int wave = tid >> 5;                  // 0..7
    const int m    = lane & 15;                 // A-matrix row (M)
    const int mc   = (m < NCLI) ? m : (m - 6);  // in-bounds column, no guard
    const int koff = (lane >> 4) << 1;          // lanes 16-31 hold K+2,K+3

    // group-segment byte offsets of this thread's 16B staging slots
    const unsigned lws0 = (unsigned)(uintptr_t)(void*)&lds[0][0] + (unsigned)tid * 16u;
    const unsigned lws1 = lws0 + (unsigned)(TILE_K * NCLI * 4);

    const int stride = (int)gridDim.x * TILE_K; // hoisted (SGPR) loop stride
    const int dFull  = d - (d % TILE_K);        // full-tile region

    v8f acc = {};                               // 16x16 f32 accumulator (8 VGPRs)

    int k0  = (int)blockIdx.x * TILE_K;
    int buf = 0;
    if (k0 < dFull)
        stage_async(lws0, x + (size_t)k0 * NCLI + tid * 4);

    for (; k0 < dFull; k0 += stride) {
        const int kn = k0 + stride;
        if (kn < dFull) {
            // prefetch next tile into the other buffer, then wait until only
            // those 5 remain outstanding: current buffer is complete.
            stage_async(buf ? lws0 : lws1, x + (size_t)kn * NCLI + tid * 4);
            asm volatile("s_wait_asynccnt 0x5" ::: "memory");
        } else {
            asm volatile("s_wait_asynccnt 0x0" ::: "memory");
        }
        __syncthreads();                        // all waves' current tile ready
        wmma_tile(lds[buf], wave, mc, koff, acc);
        __syncthreads();                        // done reading before overwrite
        buf ^= 1;
    }

    // remainder tile (owned by the block whose stride sequence hits dFull)
    if (dFull < d && k0 == dFull) {
        const int rem = (d - dFull) * NCLI;
        for (int i = tid; i < TILE_K * NCLI; i += 256)
            lds[0][i] = (i < rem) ? x[(size_t)dFull * NCLI + i] : 0.0f;
        __syncthreads();
        wmma_tile(lds[0], wave, mc, koff, acc);
        __syncthreads();
    }

    // ---- reduce 8 waves through LDS, then atomic into global G[256] ----
    float* red = &lds[0][0];                    // reuse staging LDS
#pragma unroll
    for (int v = 0; v < 8; ++v) {
        const int Mv = v + ((lane >> 4) << 3);  // C layout: vgpr v -> M=v / v+8
        red[wave * 256 + Mv * 16 + (lane & 15)] = acc[v];
    }
    __syncthreads();
    float s = 0.0f;
#pragma unroll
    for (int w = 0; w < 8; ++w) s += red[w * 256 + tid];
    atomicAdd(&G[tid], s);
}

// -------- kernel 2: all the 10x10 decision logic on one thread -------------
__global__ void select_k(const float* __restrict__ G, const float* __restrict__ r_all,
                         int* __restrict__ ii_out, float* __restrict__ xi_out) {
    if (threadIdx.x != 0) return;
    const int n = NCLI;
    float sq[NCLI], dist[NCLI][NCLI];
    for (int i = 0; i < n; ++i) sq[i] = G[i * 16 + i];
    for (int i = 0; i < n; ++i)
        for (int j = 0; j < n; ++j) {
            float d2 = sq[i] + sq[j] - 2.0f * G[i * 16 + j];
            d2 = fmaxf(d2, 0.0f);
            dist[i][j] = (d2 > 0.0f) ? sqrtf(fmaxf(d2, 1e-12f)) : 0.0f;
        }
    // per-row: 4 smallest distances (stable: strict < keeps lowest index)
    int nbh[NCLI][4]; float sum_nbh[NCLI];
    for (int i = 0; i < n; ++i) {
        bool used[NCLI];
        for (int j = 0; j < n; ++j) used[j] = false;
        float s = 0.0f;
        for (int t = 0; t < 4; ++t) {
            float best = 3.4e38f; int bj = 0;
            for (int j = 0; j < n; ++j)
                if (!used[j] && dist[i][j] < best) { best = dist[i][j]; bj = j; }
            used[bj] = true; nbh[i][t] = bj; s += best;
        }
        sum_nbh[i] = s;
    }
    int istar = 0;
    for (int i = 1; i < n; ++i) if (sum_nbh[i] < sum_nbh[istar]) istar = i;
    int ii[4]; float score[4];
    for (int t = 0; t < 4; ++t) { ii[t] = nbh[istar][t]; score[t] = sum_nbh[ii[t]]; }

    const float INV_LOG4 = 0.7213475204444817f;   // 1/ln(4)
    // MinNormalS -> p -> Es
    float mn = score[0], mx = score[0];
    for (int t = 1; t < 4; ++t) { mn = fminf(mn, score[t]); mx = fmaxf(mx, score[t]); }
    float nor[4], psum = 0.0f;
    for (int t = 0; t < 4; ++t) {
        nor[t] = (mx > mn) ? (mx - score[t]) / (mx - mn) : score[t];
        psum += nor[t];
    }
    float Es = 0.0f;
    for (int t = 0; t < 4; ++t) {
        float p = nor[t] / psum;
        if (p != 0.0f) Es += p * logf(p);
    }
    Es = -INV_LOG4 * Es;

    // reputation update: selected indices are trivially "in_set"
    float rep[4];
    for (int t = 0; t < 4; ++t) {
        float r = r_all[ii[t]];
        rep[t] = ((r > -1.0f) && (r < 1.0f)) ? (r + 0.05f) : r;
    }
    // MinNormalR -> q -> Er
    mn = rep[0]; mx = rep[0];
    for (int t = 1; t < 4; ++t) { mn = fminf(mn, rep[t]); mx = fmaxf(mx, rep[t]); }
    float norr[4], qsum = 0.0f;
    for (int t = 0; t < 4; ++t) {
        norr[t] = (mx > mn) ? (rep[t] - mn) / (mx - mn) : rep[t];
        qsum += norr[t];
    }
    float Er = 0.0f;
    for (int t = 0; t < 4; ++t) {
        float q = norr[t] / qsum;
        if (q != 0.0f) Er += q * logf(q);
    }
    Er = -INV_LOG4 * Er;

    const float denom = 2.0f - Es - Er;
    const float alpha = (1.0f - Es) / denom;
    const float beta  = (1.0f - Er) / denom;
    float w[4], wsum = 0.0f;
    for (int t = 0; t < 4; ++t) { w[t] = alpha * score[t] + beta * rep[t]; wsum += w[t]; }
    for (int t = 0; t < 4; ++t) { ii_out[t] = ii[t]; xi_out[t] = w[t] / wsum; }
}

// -------- kernel 3: out[row] = sum_t xi[t] * x[row][ii[t]] -----------------
__global__ __launch_bounds__(256) void combine_k(const float* __restrict__ x,
                                                 const int* __restrict__ ii,
                                                 const float* __restrict__ xi,
                                                 float* __restrict__ out, int d) {
    __shared__ float lds[256 * NCLI];             // 10240 B = 640 x b128
    __shared__ int sii[4];
    __shared__ float sxi[4];
    const int tid = threadIdx.x;
    if (tid < 4) { sii[tid] = ii[tid]; sxi[tid] = xi[tid]; }

    const int r0 = blockIdx.x * 256;
    if (r0 + 256 <= d) {
        const unsigned lws = (unsigned)(uintptr_t)(void*)lds + (unsigned)tid * 16u;
        const uint64_t ga = (uint64_t)(uintptr_t)(x + (size_t)r0 * NCLI)
                          + (uint64_t)tid * 16u;
        asm volatile(
            "global_load_async_to_lds_b128 %0, %1, off\n\t"
            "global_load_async_to_lds_b128 %0, %1, off offset:4096"
            :: "v"(lws), "v"(ga) : "memory");
        if (tid < 128)   // chunks 512..639 (EXEC-masked async load is legal)
            asm volatile(
                "global_load_async_to_lds_b128 %0, %1, off offset:8192"
                :: "v"(lws), "v"(ga) : "memory");
        asm volatile("s_wait_asynccnt 0x0" ::: "memory");
    } else {
        const int rem = (d - r0) * NCLI;
        for (int i = tid; i < 256 * NCLI; i += 256)
            lds[i] = (i < rem) ? x[(size_t)r0 * NCLI + i] : 0.0f;
    }
    __syncthreads();

    const int row = r0 + tid;
    if (row < d) {
        const float* p = lds + tid * NCLI;       // stride-10 words: bank-conflict-free
        out[row] = sxi[0] * p[sii[0]] + sxi[1] * p[sii[1]] +
                   sxi[2] * p[sii[2]] + sxi[3] * p[sii[3]];
    }
}

extern "C" void kernel_launch(void* const* d_in, const int* in_sizes, int n_in,
                              void* d_out, int out_size, void* d_ws, size_t ws_size,
                              hipStream_t stream) {
    const float* x     = (const float*)d_in[0];   // [1, d, 10] fp32
    const float* r_all = (const float*)d_in[1];   // [10] fp32
    float* out = (float*)d_out;                   // [1, d, 1] fp32
    const int d = in_sizes[0] / NCLI;

    float* G  = (float*)d_ws;                         // 256 floats (16x16 Gram)
    int*   ii = (int*)((char*)d_ws + 1024);           // 4 ints
    float* xi = (float*)((char*)d_ws + 1024 + 16);    // 4 floats

    zero_gram<<<1, 256, 0, stream>>>(G);

    const int nTiles = (d + TILE_K - 1) / TILE_K;
    const int gb = (GRAM_BLOCKS < nTiles) ? GRAM_BLOCKS : nTiles;
    gram_wmma<<<gb, 256, 0, stream>>>(x, G, d);

    select_k<<<1, 32, 0, stream>>>(G, r_all, ii, xi);

    combine_k<<<(d + 255) / 256, 256, 0, stream>>>(x, ii, xi, out, d);
}